// Filter_57174604645045
// MI455X (gfx1250) — compile-verified
//
#include <hip/hip_runtime.h>

typedef __attribute__((ext_vector_type(2))) float v2f;
typedef __attribute__((ext_vector_type(4))) float v4f;
typedef __attribute__((ext_vector_type(8))) float v8f;

// ---- static problem geometry (from reference) ----
#define NSAMP 32          // B*G
#define CDIM  3
#define FDIM  3277
#define TDIM  256
#define FSTRIDE   256                     // elements per f row
#define CSTRIDE   (FDIM * TDIM)           // 838912
#define SSTRIDE   (CDIM * CSTRIDE)        // 2516736 per sample
#define FC_IDX    1084
#define NOISE_LO  1638                    // rows [0,1638) are noise
#define NOISE_HI  1966                    // rows [1966,3277) are noise
#define NOISE_ROWS 2949
#define NOISE_DEN  754944.0f              // 2949 * 256
#define NCHUNK    64
#define ROWS_PER_CHUNK 47                 // 64*47 = 3008 >= 2949
#define Q_PER_ROW 64                      // 256 t / 4 (float4 per row)

// ---------------------------------------------------------------------------
// Kernel A: per-(sample, chunk) noise partial sums, b128 loads.
// Each thread streams float4s (64 threads cover one t-row; a 256-thread block
// covers 4 rows per step). Per-lane v2f accumulator pair forms a 16x4 f32
// WMMA A-fragment per wave; one V_WMMA_F32_16X16X4_F32 with B=ones folds the
// 64 lane-partials into row sums (exact f32), shfl_xor(16) completes it.
// ---------------------------------------------------------------------------
__global__ void noise_partial_kernel(const float* __restrict__ x,
                                     float* __restrict__ partials) {
    const int n     = blockIdx.y;
    const int chunk = blockIdx.x;
    const int tid   = threadIdx.x;

    int j0 = chunk * ROWS_PER_CHUNK;
    int j1 = j0 + ROWS_PER_CHUNK;
    if (j1 > NOISE_ROWS) j1 = NOISE_ROWS;
    const int nquads = (j1 - j0) * Q_PER_ROW;      // float4s in this chunk

    const v4f* xs = (const v4f*)(x + (size_t)n * SSTRIDE);

    v2f acc = {0.0f, 0.0f};
    for (int e = tid; e < nquads; e += 256) {
        int jr = e >> 6;                           // row within chunk
        int q  = e & 63;                           // float4 within row
        int j  = j0 + jr;
        int f  = (j < NOISE_LO) ? j : j + (NOISE_HI - NOISE_LO);
        size_t off = (size_t)f * (FSTRIDE / 4) + q;
        v4f a = xs[off];
        v4f b = xs[off + CSTRIDE / 4];
        v4f c = xs[off + 2 * (CSTRIDE / 4)];
        acc.x += fabsf(a.x) + fabsf(a.y) + fabsf(b.x) + fabsf(b.y)
               + fabsf(c.x) + fabsf(c.y);
        acc.y += fabsf(a.z) + fabsf(a.w) + fabsf(b.z) + fabsf(b.w)
               + fabsf(c.z) + fabsf(c.w);
    }

    // WMMA fold: D = A(16x4) * ones(4x16), row sums replicated over columns.
    v2f ones = {1.0f, 1.0f};
    v8f cz = {};
    v8f d = __builtin_amdgcn_wmma_f32_16x16x4_f32(
        /*neg_a=*/false, acc, /*neg_b=*/false, ones,
        /*c_mod=*/(short)0, cz, /*reuse_a=*/false, /*reuse_b=*/false);

    float s = d[0] + d[1] + d[2] + d[3] + d[4] + d[5] + d[6] + d[7];
    // lanes 0-15 hold rows 0-7 of a column, lanes 16-31 hold rows 8-15
    s += __shfl_xor(s, 16, 32);              // wave total (all 64 values)

    __shared__ float wsum[8];
    if ((tid & 31) == 0) wsum[tid >> 5] = s;
    __syncthreads();
    if (tid == 0) {
        float tot = 0.0f;
        for (int w = 0; w < 8; ++w) tot += wsum[w];
        partials[n * NCHUNK + chunk] = tot;
    }
}

// ---------------------------------------------------------------------------
// Kernel B: one block per sample. Reduce noise partials, build mag row at
// f=1084, first-occurrence argmax, windowed signal sum, SNR -> half_band.
// ---------------------------------------------------------------------------
__global__ void stats_kernel(const float* __restrict__ x,
                             const float* __restrict__ partials,
                             int* __restrict__ mid_out,
                             int* __restrict__ hb_out) {
    const int n   = blockIdx.x;
    const int tid = threadIdx.x;

    __shared__ float red[256];
    __shared__ float mag[256];

    red[tid] = (tid < NCHUNK) ? partials[n * NCHUNK + tid] : 0.0f;
    __syncthreads();
    for (int s = 128; s > 0; s >>= 1) {
        if (tid < s) red[tid] += red[tid + s];
        __syncthreads();
    }

    size_t base = (size_t)n * SSTRIDE + (size_t)FC_IDX * FSTRIDE + tid;
    mag[tid] = fabsf(x[base]) + fabsf(x[base + CSTRIDE]) +
               fabsf(x[base + 2 * CSTRIDE]);
    __syncthreads();

    if (tid == 0) {
        // first-occurrence argmax (matches jnp.argmax tie-break)
        float best = mag[0]; int mid = 0;
        for (int t = 1; t < TDIM; ++t)
            if (mag[t] > best) { best = mag[t]; mid = t; }

        int lo = mid - 8; if (lo < 0) lo = 0;
        int hi = mid + 8; if (hi > TDIM) hi = TDIM;
        float sig = 0.0f;
        for (int t = lo; t < hi; ++t) sig += mag[t];

        float noise = red[0] / NOISE_DEN;
        float diff  = sig - noise;
        float snr   = 10.0f * log10f((diff * diff) / (noise * noise));
        int hb = (int)(12.5f * (snr - 20.0f) + 27.0f);   // trunc toward zero
        if (hb < 8) hb = 8;

        mid_out[n] = mid;
        hb_out[n]  = hb;
    }
}

// ---------------------------------------------------------------------------
// Kernel C: H[t] = max over samples covering t of half_band (0 if none).
// Union of centered f-intervals == single interval of max half-width.
// ---------------------------------------------------------------------------
__global__ void hmask_kernel(const int* __restrict__ mid,
                             const int* __restrict__ hb,
                             int* __restrict__ H) {
    const int t = threadIdx.x;
    int h = 0;
    for (int n = 0; n < NSAMP; ++n) {
        int m = mid[n];
        if (t >= m - 8 && t < m + 8) {
            int v = hb[n];
            h = (v > h) ? v : h;
        }
    }
    H[t] = h;
}

// ---------------------------------------------------------------------------
// Kernel D: streaming masked copy, float4 per thread, nontemporal (no reuse).
// out[.,f,t] = (1084-H[t] <= f < 1084+H[t]) ? x : 0
// ---------------------------------------------------------------------------
__global__ void mask_copy_kernel(const float* __restrict__ x,
                                 float* __restrict__ out,
                                 const int* __restrict__ H) {
    __shared__ int sH[TDIM];
    const int tid = threadIdx.x;
    sH[tid] = H[tid];
    __syncthreads();

    long long v   = (long long)blockIdx.x * 256 + tid;   // float4 index
    int  t4  = (int)(v & 63);
    long long row = v >> 6;                              // (n*C + c)*F + f
    int  f   = (int)(row % FDIM);

    const v4f* x4 = (const v4f*)x;
    v4f*       o4 = (v4f*)out;

    v4f xi = __builtin_nontemporal_load(&x4[v]);
    int tt = t4 * 4;
    v4f r;
    int h;
    h = sH[tt + 0]; r.x = (f >= FC_IDX - h && f < FC_IDX + h) ? xi.x : 0.0f;
    h = sH[tt + 1]; r.y = (f >= FC_IDX - h && f < FC_IDX + h) ? xi.y : 0.0f;
    h = sH[tt + 2]; r.z = (f >= FC_IDX - h && f < FC_IDX + h) ? xi.z : 0.0f;
    h = sH[tt + 3]; r.w = (f >= FC_IDX - h && f < FC_IDX + h) ? xi.w : 0.0f;
    __builtin_nontemporal_store(r, &o4[v]);
}

// ---------------------------------------------------------------------------
extern "C" void kernel_launch(void* const* d_in, const int* in_sizes, int n_in,
                              void* d_out, int out_size, void* d_ws, size_t ws_size,
                              hipStream_t stream) {
    const float* x  = (const float*)d_in[0];
    float* out      = (float*)d_out;

    // workspace layout (9.5 KB): partials | mid | hb | H
    float* ws_noise = (float*)d_ws;                     // 32*64 floats
    int*   ws_mid   = (int*)(ws_noise + NSAMP * NCHUNK);// 32 ints
    int*   ws_hb    = ws_mid + NSAMP;                   // 32 ints
    int*   ws_H     = ws_hb + NSAMP;                    // 256 ints (16B aligned)

    noise_partial_kernel<<<dim3(NCHUNK, NSAMP), 256, 0, stream>>>(x, ws_noise);
    stats_kernel<<<NSAMP, 256, 0, stream>>>(x, ws_noise, ws_mid, ws_hb);
    hmask_kernel<<<1, 256, 0, stream>>>(ws_mid, ws_hb, ws_H);

    // 80,535,552 elements / 4 per thread / 256 threads = 78,648 blocks exactly
    mask_copy_kernel<<<78648, 256, 0, stream>>>(x, out, ws_H);
}